// RotateSoftNMS_18854906429936
// MI455X (gfx1250) — compile-verified
//
#include <hip/hip_runtime.h>
#include <hip/hip_bf16.h>
#include <math.h>

#define NMS_THRESH 0.3f
#define SCORE_THRESH 0.001f
#define EPS 1e-9f
#define N_MAX 2048

typedef unsigned int u32;
typedef u32 __attribute__((ext_vector_type(4))) v4u;
typedef int  __attribute__((ext_vector_type(4))) v4i;
typedef int  __attribute__((ext_vector_type(8))) v8i;

#if defined(__has_builtin)
#if __has_builtin(__builtin_amdgcn_tensor_load_to_lds) && __has_builtin(__builtin_amdgcn_s_wait_tensorcnt)
#define USE_TDM 1
#endif
#endif
#ifndef USE_TDM
#define USE_TDM 0
#endif

// ---------------- Phase 0: corners + areas ----------------
__global__ __launch_bounds__(256) void rsnms_prep(const float* __restrict__ boxes,
                                                  float* __restrict__ corn,
                                                  float* __restrict__ areas, int n) {
  int i = blockIdx.x * blockDim.x + threadIdx.x;
  if (i >= n) return;
  float xc = boxes[i * 5 + 0], yc = boxes[i * 5 + 1];
  float w  = boxes[i * 5 + 2], h  = boxes[i * 5 + 3];
  float t  = boxes[i * 5 + 4] * 0.017453292519943295f;
  float c = cosf(t), s = sinf(t);
  float w2 = 0.5f * w, h2 = 0.5f * h;
  const float lx[4] = {-w2,  w2, w2, -w2};
  const float ly[4] = {-h2, -h2, h2,  h2};
#pragma unroll
  for (int k = 0; k < 4; ++k) {
    corn[i * 8 + k]     = xc + lx[k] * c - ly[k] * s;   // x, CCW order
    corn[i * 8 + 4 + k] = yc + lx[k] * s + ly[k] * c;   // y
  }
  areas[i] = w * h;
}

// ---------------- Phase 1: dense IoU matrix (Sutherland-Hodgman clip) -------
__global__ __launch_bounds__(256) void rsnms_iou(const float* __restrict__ corn,
                                                 const float* __restrict__ areas,
                                                 float* __restrict__ iouM, int n) {
  int j = blockIdx.x * blockDim.x + threadIdx.x;
  int i = blockIdx.y;                 // uniform per block -> scalar loads of A
  if (j >= n) return;
  float ax[4], ay[4];
  float px[8], py[8];
#pragma unroll
  for (int k = 0; k < 4; ++k) {
    ax[k] = corn[i * 8 + k];
    ay[k] = corn[i * 8 + 4 + k];
    px[k] = corn[(size_t)j * 8 + k];
    py[k] = corn[(size_t)j * 8 + 4 + k];
  }
  int cnt = 4;
  // clip quad j by the 4 half-planes of quad i (both CCW; inside = left side)
#pragma unroll
  for (int e = 0; e < 4; ++e) {
    float x1 = ax[e], y1 = ay[e];
    float x2 = ax[(e + 1) & 3], y2 = ay[(e + 1) & 3];
    float ex = x2 - x1, ey = y2 - y1;
    float d[8], qx[8], qy[8];
    for (int k = 0; k < cnt; ++k) d[k] = ex * (py[k] - y1) - ey * (px[k] - x1);
    int m = 0;
    for (int k = 0; k < cnt; ++k) {
      int kn = (k + 1 == cnt) ? 0 : k + 1;
      bool ins_s = d[k] >= 0.0f, ins_t = d[kn] >= 0.0f;
      if (ins_s) { qx[m] = px[k]; qy[m] = py[k]; ++m; }
      if (ins_s != ins_t) {
        float r = d[k] / (d[k] - d[kn]);
        qx[m] = px[k] + r * (px[kn] - px[k]);
        qy[m] = py[k] + r * (py[kn] - py[k]);
        ++m;
      }
    }
    cnt = m;
    for (int k = 0; k < cnt; ++k) { px[k] = qx[k]; py[k] = qy[k]; }
    if (cnt == 0) break;
  }
  float a2 = 0.0f;
  for (int k = 0; k < cnt; ++k) {
    int kn = (k + 1 == cnt) ? 0 : k + 1;
    a2 += px[k] * py[kn] - px[kn] * py[k];
  }
  float inter = 0.5f * fabsf(a2);
  iouM[(size_t)i * n + j] = inter / (areas[i] + areas[j] - inter + EPS);
}

// ---------------- Phase 2: serial soft-NMS scan (single workgroup) ----------
__global__ __launch_bounds__(1024) void rsnms_scan(const float* __restrict__ scores_in,
                                                   const float* __restrict__ iouM,
                                                   float* __restrict__ out, int n) {
  __shared__ float s_sc[N_MAX];
  __shared__ int   s_sel[N_MAX];
  __shared__ float s_row[N_MAX];
  __shared__ float s_wv[32];
  __shared__ int   s_wi[32];
  __shared__ int   s_bidx;
  __shared__ float s_bval;

  const int t = threadIdx.x;
  const int nth = blockDim.x;
  const int lane = t & 31, wid = t >> 5;

  for (int j = t; j < n; j += nth) { s_sc[j] = scores_in[j]; s_sel[j] = 0; }
  __syncthreads();

  for (int it = 0; it < n; ++it) {
    // --- argmax over avail = sel ? -inf : score (first-index tiebreak) ---
    float bv = -INFINITY; int bi = 0;
    for (int j = t; j < n; j += nth) {
      float v = s_sel[j] ? -INFINITY : s_sc[j];
      if (v > bv) { bv = v; bi = j; }
    }
#pragma unroll
    for (int off = 16; off; off >>= 1) {
      float ov = __shfl_xor(bv, off);
      int   oi = __shfl_xor(bi, off);
      if (ov > bv || (ov == bv && oi < bi)) { bv = ov; bi = oi; }
    }
    if (lane == 0) { s_wv[wid] = bv; s_wi[wid] = bi; }
    __syncthreads();

    if (wid == 0) {
      const int nw = nth >> 5;
      float v = (lane < nw) ? s_wv[lane] : -INFINITY;
      int  ix = (lane < nw) ? s_wi[lane] : 0x7fffffff;
#pragma unroll
      for (int off = 16; off; off >>= 1) {
        float ov = __shfl_xor(v, off);
        int   oi = __shfl_xor(ix, off);
        if (ov > v || (ov == v && oi < ix)) { v = ov; ix = oi; }
      }
      if (lane == 0) { s_bidx = ix; s_bval = v; }
#if USE_TDM
      // TDM: DMA row `ix` (n floats) from L2-resident IoU matrix into LDS.
      {
        u32 row = (u32)__builtin_amdgcn_readfirstlane(ix);
        unsigned long long ga =
            (unsigned long long)(uintptr_t)(iouM + (size_t)row * (size_t)n);
        u32 lds = (u32)(uintptr_t)(&s_row[0]);   // low 32 bits = LDS byte offset
        v4u g0;
        g0.x = 1u;                                // count=1, user descriptor
        g0.y = lds;                               // lds_addr
        g0.z = (u32)(ga & 0xffffffffu);           // global_addr[31:0]
        g0.w = (u32)((ga >> 32) & 0x1ffffffu) | (2u << 30);  // addr[56:32] | type=2
        v8i g1;
        g1[0] = (int)(2u << 16);                  // data_size=4B; no mask/pad/iter
        g1[1] = (int)(((u32)n & 0xffffu) << 16);  // tensor_dim0[15:0] @ bits 63:48
        g1[2] = (int)(((u32)n >> 16) | (1u << 16)); // dim0 hi | tensor_dim1=1
        g1[3] = (int)(((u32)n & 0xffffu) << 16);  // tile_dim0 = n @ bits 127:112
        g1[4] = 0;                                // tile_dim1/2 unused -> 1-D tile
        g1[5] = (int)(u32)n;                      // tensor_dim0_stride[31:0]
        g1[6] = 0;
        g1[7] = 0;
        v4i gz4 = {0, 0, 0, 0};
        v8i gz8 = {0, 0, 0, 0, 0, 0, 0, 0};
        // 6-arg form (clang-23 / therock-10.0 headers):
        // (g0: u32x4, g1: i32x8, g2: i32x4, g3: i32x4, g4: i32x8, cpol: i32)
        __builtin_amdgcn_tensor_load_to_lds(g0, g1, gz4, gz4, gz8, 0);
        __builtin_amdgcn_s_wait_tensorcnt(0);
      }
#endif
    }
    __syncthreads();

    const int  bidx = s_bidx;
    const float bval = s_bval;
    const bool valid = bval > SCORE_THRESH;

    for (int j = t; j < n; j += nth) {
#if USE_TDM
      float vio = s_row[j];
#else
      float vio = iouM[(size_t)bidx * n + j];
#endif
      float w = (vio > NMS_THRESH) ? (1.0f - vio) : 1.0f;   // METHOD==1 linear
      bool is_i = (j == bidx);
      if (is_i || s_sel[j] || !valid) w = 1.0f;
      s_sc[j] *= w;
      if (is_i) s_sel[j] = 1;
    }
    if (t == 0) {
      out[it]         = (float)bidx;                         // order
      out[n + it]     = (bval > SCORE_THRESH) ? 1.0f : 0.0f; // keep
      out[2 * n + it] = bval;                                // rec
    }
    __syncthreads();
  }
}

extern "C" void kernel_launch(void* const* d_in, const int* in_sizes, int n_in,
                              void* d_out, int out_size, void* d_ws, size_t ws_size,
                              hipStream_t stream) {
  const float* boxes  = (const float*)d_in[0];
  const float* scores = (const float*)d_in[1];
  const int n = in_sizes[1];          // N from scores

  float* corn  = (float*)d_ws;                    // n*8 floats
  float* areas = corn + (size_t)n * 8;            // n floats
  size_t off   = (((size_t)n * 9) + 3) & ~(size_t)3;
  float* iouM  = corn + off;                      // n*n floats (16 MB @ n=2048)
  float* out   = (float*)d_out;

  dim3 bp(256), gp((n + 255) / 256);
  rsnms_prep<<<gp, bp, 0, stream>>>(boxes, corn, areas, n);

  dim3 bi(256), gi((n + 255) / 256, n);
  rsnms_iou<<<gi, bi, 0, stream>>>(corn, areas, iouM, n);

  rsnms_scan<<<1, 1024, 0, stream>>>(scores, iouM, out, n);
}